// arrayPC_naive_62294205662030
// MI455X (gfx1250) — compile-verified
//
#include <hip/hip_runtime.h>
#include <math.h>
#include <stdint.h>

#define NPOS 96      // n
#define KST  64      // k
#define NW   95      // n-1
#define KW   63      // k-1
#define NPAIR (NW * KW)   // 5985 weight pairs
#define BLK  256

typedef float v2f __attribute__((ext_vector_type(2)));
typedef float v8f __attribute__((ext_vector_type(8)));

// ---------------------------------------------------------------------------
// Kernel 1: W_full = softmax(W, axis=2) * A1 + A2 (masked), stored PLANAR:
//   ws[0      .. NPAIR-1] = w0 plane (shift weights)
//   ws[NPAIR  .. 2*NPAIR-1] = w1 plane (stay weights)
//   ws[2*NPAIR .. +63]    = softmax(endW)
// ---------------------------------------------------------------------------
__global__ __launch_bounds__(256) void prep_kernel(
    const float* __restrict__ W, const float* __restrict__ endW,
    float* __restrict__ wf, float* __restrict__ e) {
  int gid = blockIdx.x * blockDim.x + threadIdx.x;
  if (gid < NPAIR) {
    int i = gid / KW;           // weight row (0..94)
    int j = gid % KW;           // state column (0..62)
    float w0 = W[gid * 2 + 0];
    float w1 = W[gid * 2 + 1];
    float mx = fmaxf(w0, w1);
    float e0 = expf(w0 - mx), e1 = expf(w1 - mx);
    float s = e0 + e1;
    float r0 = e0 / s, r1 = e1 / s;
    if (i < KST - 2) {          // rows 0..61 get masked/overridden
      if (j == i + 1)      { r0 = 1.0f; r1 = 0.0f; }
      else if (j > i + 1)  { r0 = 0.0f; r1 = 0.0f; }
      // j <= i: keep softmax
    }
    wf[gid]         = r0;       // plane 0 (x_i == 1 path)
    wf[NPAIR + gid] = r1;       // plane 1 (x_i == 0 path)
  } else if (gid == NPAIR) {
    // endW softmax (64 values) — trivial, single thread
    float mx = -3.0e38f;
    for (int j = 0; j < KST; ++j) mx = fmaxf(mx, endW[j]);
    float s = 0.0f;
    for (int j = 0; j < KST; ++j) s += expf(endW[j] - mx);
    for (int j = 0; j < KST; ++j) e[j] = expf(endW[j] - mx) / s;
  }
}

// ---------------------------------------------------------------------------
// Kernel 2: per-row scan with F[64] in VGPRs + WMMA dot-product epilogue.
// ---------------------------------------------------------------------------
__global__ __launch_bounds__(BLK) void scan_kernel(
    const int* __restrict__ x, const float* __restrict__ wf_g,
    const float* __restrict__ e_g, float* __restrict__ out) {
  // Weight planes (11970 floats, 48 KB); later reused as per-wave F staging.
  __shared__ float ldsbuf[NPAIR * 2];
  __shared__ float e[KST];

  const int tid  = threadIdx.x;
  const int lane = tid & 31;
  const int wv   = tid >> 5;
  const int b    = blockIdx.x * BLK + tid;

  // ---- Stage weights into LDS (async global->LDS; path proven active) ----
#if __has_builtin(__builtin_amdgcn_global_load_async_to_lds_b32)
  {
    unsigned lbase = (unsigned)(uintptr_t)&ldsbuf[0];
    for (int idx = tid; idx < NPAIR * 2; idx += BLK) {
      __builtin_amdgcn_global_load_async_to_lds_b32(
          (__attribute__((address_space(1))) int*)(uintptr_t)(wf_g + idx),
          (__attribute__((address_space(3))) int*)(lbase + (unsigned)idx * 4u),
          0, 0);
    }
#if __has_builtin(__builtin_amdgcn_s_wait_asynccnt)
    __builtin_amdgcn_s_wait_asynccnt(0);
#else
    asm volatile("s_wait_asynccnt 0" ::: "memory");
#endif
  }
#else
  for (int idx = tid; idx < NPAIR * 2; idx += BLK) ldsbuf[idx] = wf_g[idx];
#endif
  if (tid < KST) e[tid] = e_g[tid];
  __syncthreads();

  // ---- Pack this row's 96 x-bits into 3 registers (24 x int4 loads) ----
  const int4* xrow = (const int4*)(x + (size_t)b * NPOS);
  unsigned mw[3] = {0u, 0u, 0u};
#pragma unroll
  for (int q = 0; q < 24; ++q) {
    int4 v = xrow[q];
    const int base = q * 4;                  // constant per unrolled iter
    mw[base >> 5] |= ((unsigned)(v.x & 1) << ((base + 0) & 31)) |
                     ((unsigned)(v.y & 1) << ((base + 1) & 31)) |
                     ((unsigned)(v.z & 1) << ((base + 2) & 31)) |
                     ((unsigned)(v.w & 1) << ((base + 3) & 31));
  }

  // ---- Init F (position 0) ----
  float x0f = (float)(mw[0] & 1u);
  float p = 1.0f - x0f;                      // cumprod(1-x) at pos 0
  float F[KST];
  F[0] = p;
  F[1] = x0f;
#pragma unroll
  for (int j = 2; j < KST; ++j) F[j] = 0.0f;

  // ---- 95-step scan ----
  // Plane base selected ONCE per step by x_i (one cndmask); then per element:
  // one ds_load_b32 (2-group broadcast) + one cndmask (shift/stay src) + mul.
#pragma unroll
  for (int seg = 0; seg < 3; ++seg) {        // unrolled so mw[seg] is constant-indexed
    const unsigned bits = mw[seg];
    const int tstart = (seg == 0) ? 1 : 0;
    for (int t = tstart; t < 32; ++t) {
      const int i = seg * 32 + t;            // position 1..95
      const bool xi = (bits >> t) & 1u;
      const float* wrow = &ldsbuf[(xi ? 0 : NPAIR) + (i - 1) * KW];
#pragma unroll
      for (int j = KST - 1; j >= 1; --j) {   // descending: F[j-1] is prev-step
        float w = wrow[j - 1];
        float src = xi ? F[j - 1] : F[j];
        F[j] = w * src;
      }
      p = xi ? 0.0f : p;
      F[0] = p;
    }
  }

  // ---- Epilogue: out[b] = log(dot(F, e)) via v_wmma_f32_16x16x4_f32 ----
#if __has_builtin(__builtin_amdgcn_wmma_f32_16x16x4_f32)
  // Per-wave: stage 16 rows of F into LDS, compute 16-row dot via 16 chained
  // WMMAs (B = e chunk replicated across all 16 columns, so every column of D
  // equals dot(F_row, e)).
  float* buf = &ldsbuf[wv * (16 * 68)];      // padded stride 68 -> no bank conflicts
  const int m = lane & 15;                   // A-matrix row M
  const int off = (lane < 16) ? 0 : 2;       // f32 A layout: lanes16-31 hold K+2
#pragma unroll 1
  for (int r = 0; r < 2; ++r) {              // two 16-row groups per wave
    __syncthreads();                          // also separates weight reads from reuse
    if ((lane >> 4) == r) {
#pragma unroll
      for (int j = 0; j < KST; ++j) buf[m * 68 + j] = F[j];
    }
    __syncthreads();
    v8f acc = {};
#pragma unroll
    for (int kc = 0; kc < 16; ++kc) {        // K = 64 in chunks of 4
      v2f a, bb;
      a.x  = buf[m * 68 + kc * 4 + off + 0]; // VGPR0: K0 / K2
      a.y  = buf[m * 68 + kc * 4 + off + 1]; // VGPR1: K1 / K3
      bb.x = e[kc * 4 + off + 0];            // B rows replicated over N
      bb.y = e[kc * 4 + off + 1];
      acc = __builtin_amdgcn_wmma_f32_16x16x4_f32(
          false, a, false, bb, (short)0, acc, false, false);
    }
    // D layout: lane0 VGPR v = D[v][0] (rows 0..7); lane16 VGPR v = D[v+8][0].
    const int rowBase = blockIdx.x * BLK + wv * 32 + r * 16;
    if (lane == 0) {
#pragma unroll
      for (int v = 0; v < 8; ++v) out[rowBase + v] = logf(acc[v]);
    } else if (lane == 16) {
#pragma unroll
      for (int v = 0; v < 8; ++v) out[rowBase + 8 + v] = logf(acc[v]);
    }
  }
#else
  float acc = 0.0f;
#pragma unroll
  for (int j = 0; j < KST; ++j) acc += F[j] * e[j];
  out[b] = logf(acc);
#endif
}

// ---------------------------------------------------------------------------
extern "C" void kernel_launch(void* const* d_in, const int* in_sizes, int n_in,
                              void* d_out, int out_size, void* d_ws, size_t ws_size,
                              hipStream_t stream) {
  const int*   x    = (const int*)d_in[0];     // bool x -> int per harness rules
  const float* W    = (const float*)d_in[1];   // (95, 63, 2) f32
  const float* endW = (const float*)d_in[2];   // (64, 1) f32
  float* out = (float*)d_out;

  float* wf = (float*)d_ws;                    // 2*NPAIR floats: planar W_full
  float* e  = wf + NPAIR * 2;                  // 64 floats: softmax(endW)

  const int nB = in_sizes[0] / NPOS;           // 131072 batch rows

  prep_kernel<<<(NPAIR + 256) / 256, 256, 0, stream>>>(W, endW, wf, e);
  scan_kernel<<<nB / BLK, BLK, 0, stream>>>(x, wf, e, out);
}